// DataFlowGraphEncoder_55490977465025
// MI455X (gfx1250) — compile-verified
//
#include <hip/hip_runtime.h>
#include <math.h>

typedef __attribute__((ext_vector_type(2))) float v2f;
typedef __attribute__((ext_vector_type(8))) float v8f;

#define NNODES 16384
#define NPER   512
#define NGRAPH 32
#define NEDGE  131072
#define HDIM   128

// ---------------------------------------------------------------- helpers
__device__ __forceinline__ void atomicMaxF(float* addr, float val){
  // ordered-int trick: works for mixed-sign floats with init = most-negative
  if (val >= 0.f) atomicMax((int*)addr, __float_as_int(val));
  else            atomicMin((unsigned int*)addr, __float_as_uint(val));
}

__global__ void k_fill(float* __restrict__ p, float v, int n){
  int t = blockIdx.x*blockDim.x + threadIdx.x;
  if (t < n) p[t] = v;
}

// ---------------------------------------------------------------- WMMA GEMM (f32)
// C[M,N] = A[M,K] @ Bop + bias.  Bop = TRANSB ? B^T (B stored [N,K]) : B ([K,N]).
// grid = (N/64, M/128), block = 256 threads (8 waves). Wave -> 16x64 tile.
// Uses V_WMMA_F32_16X16X4_F32; all waves compute full tiles (EXEC all ones).
template<bool TRANSB>
__global__ void __launch_bounds__(256)
k_gemm(const float* __restrict__ A, const float* __restrict__ B,
       const float* __restrict__ bias, float* __restrict__ C,
       int M, int N, int K)
{
  const int lane = threadIdx.x & 31;
  const int wave = threadIdx.x >> 5;
  const int m0 = blockIdx.y*128 + wave*16;
  const int n0 = blockIdx.x*64;
  const int mr = lane & 15;            // A row within tile
  const int nr = lane & 15;            // C/B col within 16
  const int kh = (lane >> 4) << 1;     // K offset 0/2 within group of 4

  v8f acc0 = {}; v8f acc1 = {}; v8f acc2 = {}; v8f acc3 = {};
  const float* Arow = A + (size_t)(m0 + mr)*K + kh;

  for (int k0 = 0; k0 < K; k0 += 4){
    v2f a = *(const v2f*)(Arow + k0);
    v2f b0, b1, b2, b3;
    if (TRANSB){
      const float* Bb = B + k0 + kh;
      b0 = *(const v2f*)(Bb + (size_t)(n0 +  0 + nr)*K);
      b1 = *(const v2f*)(Bb + (size_t)(n0 + 16 + nr)*K);
      b2 = *(const v2f*)(Bb + (size_t)(n0 + 32 + nr)*K);
      b3 = *(const v2f*)(Bb + (size_t)(n0 + 48 + nr)*K);
    } else {
      const float* Br0 = B + (size_t)(k0 + kh)*N + n0 + nr;
      const float* Br1 = Br0 + N;
      b0.x = Br0[ 0]; b0.y = Br1[ 0];
      b1.x = Br0[16]; b1.y = Br1[16];
      b2.x = Br0[32]; b2.y = Br1[32];
      b3.x = Br0[48]; b3.y = Br1[48];
    }
    acc0 = __builtin_amdgcn_wmma_f32_16x16x4_f32(false, a, false, b0, (short)0, acc0, false, false);
    acc1 = __builtin_amdgcn_wmma_f32_16x16x4_f32(false, a, false, b1, (short)0, acc1, false, false);
    acc2 = __builtin_amdgcn_wmma_f32_16x16x4_f32(false, a, false, b2, (short)0, acc2, false, false);
    acc3 = __builtin_amdgcn_wmma_f32_16x16x4_f32(false, a, false, b3, (short)0, acc3, false, false);
  }

  const int mh = (lane >> 4) << 3;     // C row offset 0/8
  float bb0 = bias ? bias[n0 +  0 + nr] : 0.f;
  float bb1 = bias ? bias[n0 + 16 + nr] : 0.f;
  float bb2 = bias ? bias[n0 + 32 + nr] : 0.f;
  float bb3 = bias ? bias[n0 + 48 + nr] : 0.f;
  #pragma unroll
  for (int r = 0; r < 8; ++r){
    float* Cr = C + (size_t)(m0 + mh + r)*N + n0 + nr;
    Cr[ 0] = acc0[r] + bb0;
    Cr[16] = acc1[r] + bb1;
    Cr[32] = acc2[r] + bb2;
    Cr[48] = acc3[r] + bb3;
  }
}

// ---------------------------------------------------------------- embeddings
// x[i,j] += var_tab[vt[i]][j%32] + op_tab[ot[i]][j%32]   (H/4 tables tiled x4)
__global__ void k_add_embed(float* __restrict__ x, const int* __restrict__ vt,
                            const int* __restrict__ ot, const float* __restrict__ vtab,
                            const float* __restrict__ otab){
  int t = blockIdx.x*blockDim.x + threadIdx.x;   // NNODES*HDIM
  int i = t >> 7, j = t & 127;
  x[t] += vtab[vt[i]*32 + (j & 31)] + otab[ot[i]*32 + (j & 31)];
}

// ---------------------------------------------------------------- edge attention
__global__ void k_edge_scores(const float* __restrict__ Q, const float* __restrict__ Kp,
                              const int* __restrict__ src, const int* __restrict__ dst,
                              float* __restrict__ sc, float* __restrict__ mbuf){
  int t = blockIdx.x*blockDim.x + threadIdx.x;   // NEDGE*4
  int e = t >> 2, h = t & 3;
  int s = src[e], d = dst[e];
  const float4* qp = (const float4*)(Q  + (size_t)d*512 + h*128);
  const float4* kp = (const float4*)(Kp + (size_t)s*512 + h*128);
  float acc = 0.f;
  #pragma unroll 8
  for (int i = 0; i < 32; ++i){
    float4 qa = qp[i], ka = kp[i];
    acc += qa.x*ka.x + qa.y*ka.y + qa.z*ka.z + qa.w*ka.w;
  }
  float v = acc * 0.0883883476483184f;  // 1/sqrt(128)
  sc[t] = v;
  atomicMaxF(mbuf + (size_t)d*4 + h, v);
}

__global__ void k_edge_exp(const int* __restrict__ dst, const float* __restrict__ mbuf,
                           float* __restrict__ sc, float* __restrict__ den){
  int t = blockIdx.x*blockDim.x + threadIdx.x;   // NEDGE*4
  int e = t >> 2, h = t & 3;
  int d = dst[e];
  float ex = expf(sc[t] - mbuf[(size_t)d*4 + h]);
  sc[t] = ex;
  atomicAdd(den + (size_t)d*4 + h, ex);
}

__global__ void k_edge_alpha(const int* __restrict__ dst, const float* __restrict__ den,
                             float* __restrict__ sc){
  int t = blockIdx.x*blockDim.x + threadIdx.x;   // NEDGE*4
  int e = t >> 2, h = t & 3;
  int d = dst[e];
  sc[t] = sc[t] / (den[(size_t)d*4 + h] + 1e-16f);
}

// agg[dst,d] += 0.25 * sum_h alpha[e,h] * V[src, h*128+d]   (head-mean folded in)
__global__ void k_edge_agg(const int* __restrict__ src, const int* __restrict__ dst,
                           const float* __restrict__ alpha, const float* __restrict__ V,
                           float* __restrict__ agg){
  int t = blockIdx.x*blockDim.x + threadIdx.x;   // NEDGE*128
  int e = t >> 7, d = t & 127;
  int s = src[e], dn = dst[e];
  const float* vp = V + (size_t)s*512 + d;
  const float* ap = alpha + (size_t)e*4;
  float sum = ap[0]*vp[0] + ap[1]*vp[128] + ap[2]*vp[256] + ap[3]*vp[384];
  atomicAdd(agg + (size_t)dn*128 + d, 0.25f*sum);
}

// x = gelu(agg + skip)   (exact gelu)
__global__ void k_conv_finish(const float* __restrict__ agg, const float* __restrict__ skip,
                              float* __restrict__ x){
  int t = blockIdx.x*blockDim.x + threadIdx.x;
  float v = agg[t] + skip[t];
  x[t] = 0.5f*v*(1.f + erff(v*0.7071067811865475f));
}

// ---------------------------------------------------------------- GRU (sequential scan)
// one block per graph; threads 0..383 each own one row of Whh; h/gh in LDS
__global__ void __launch_bounds__(384)
k_gru(const float* __restrict__ gi, const float* __restrict__ Whh,
      const float* __restrict__ bhh, float* __restrict__ hs){
  __shared__ float h[HDIM];
  __shared__ float gh[3*HDIM];
  const int b = blockIdx.x, tid = threadIdx.x;
  if (tid < HDIM) h[tid] = 0.f;
  __syncthreads();
  const float* w  = Whh + (size_t)tid*HDIM;
  const float wb  = bhh[tid];
  for (int t = 0; t < NPER; ++t){
    float acc = wb;
    #pragma unroll 8
    for (int i = 0; i < HDIM; ++i) acc += w[i]*h[i];
    gh[tid] = acc;
    __syncthreads();
    if (tid < HDIM){
      const float* g = gi + ((size_t)b*NPER + t)*(3*HDIM);
      float r  = 1.f/(1.f + expf(-(g[tid]        + gh[tid])));
      float z  = 1.f/(1.f + expf(-(g[HDIM+tid]   + gh[HDIM+tid])));
      float nn = tanhf(g[2*HDIM+tid] + r*gh[2*HDIM+tid]);
      float hn = (1.f - z)*nn + z*h[tid];
      hs[((size_t)b*NPER + t)*HDIM + tid] = hn;
      h[tid] = hn;
    }
    __syncthreads();
  }
}

// ---------------------------------------------------------------- taint logits
__global__ void k_taint_logits(const float* __restrict__ tf, const float* __restrict__ Wc,
                               const float* __restrict__ bc, float* __restrict__ out){
  int t = blockIdx.x*blockDim.x + threadIdx.x;   // NNODES*2
  int i = t >> 1, c = t & 1;
  const float* f = tf + (size_t)i*HDIM;
  float acc = bc[c];
  #pragma unroll 8
  for (int j = 0; j < HDIM; ++j) acc += f[j]*Wc[j*2 + c];
  out[t] = acc;
}

// ---------------------------------------------------------------- MHA (flash-style)
// grid (graph, head, rowblock), block 128; one query row per thread, dh=32
__global__ void __launch_bounds__(128)
k_mha(const float* __restrict__ QKV, float* __restrict__ O){
  const int b = blockIdx.x, h = blockIdx.y;
  const int q = blockIdx.z*128 + threadIdx.x;
  const float* base = QKV + (size_t)b*NPER*384;
  float qv[32];
  const float* qp = base + (size_t)q*384 + h*32;
  #pragma unroll
  for (int i = 0; i < 32; ++i) qv[i] = qp[i];
  float m = -3.4e38f, l = 0.f;
  float acc[32];
  #pragma unroll
  for (int i = 0; i < 32; ++i) acc[i] = 0.f;
  for (int j = 0; j < NPER; ++j){
    const float* kp = base + (size_t)j*384 + 128 + h*32;
    float s = 0.f;
    #pragma unroll
    for (int i = 0; i < 32; ++i) s += qv[i]*kp[i];
    s *= 0.1767766952966369f;           // 1/sqrt(32)
    float mn = fmaxf(m, s);
    float cs = expf(m - mn);
    float p  = expf(s - mn);
    l = l*cs + p;
    const float* vp = kp + 128;
    #pragma unroll
    for (int i = 0; i < 32; ++i) acc[i] = acc[i]*cs + p*vp[i];
    m = mn;
  }
  float inv = 1.f/l;
  float* op = O + ((size_t)b*NPER + q)*HDIM + h*32;
  #pragma unroll
  for (int i = 0; i < 32; ++i) op[i] = acc[i]*inv;
}

// ---------------------------------------------------------------- outputs
__global__ void k_combine(const float* __restrict__ x, const float* __restrict__ tf,
                          const float* __restrict__ ud, float* __restrict__ out){
  int t = blockIdx.x*blockDim.x + threadIdx.x;
  out[t] = x[t] + tf[t] + ud[t];
}

__global__ void k_graph_mean(const float* __restrict__ comb, float* __restrict__ out){
  int b = blockIdx.x, j = threadIdx.x;   // 32 blocks x 128
  const float* p = comb + (size_t)b*NPER*HDIM + j;
  float s = 0.f;
  for (int t = 0; t < NPER; ++t) s += p[(size_t)t*HDIM];
  out[b*HDIM + j] = s * (1.f/NPER);
}

__global__ void __launch_bounds__(512)
k_taint_graph(const float* __restrict__ logits, float* __restrict__ out){
  __shared__ float s0[512];
  __shared__ float s1[512];
  int b = blockIdx.x, t = threadIdx.x;   // 32 blocks x 512
  const float* l = logits + ((size_t)b*NPER + t)*2;
  float a = l[0], c = l[1];
  float mx = fmaxf(a, c);
  float e0 = expf(a - mx), e1 = expf(c - mx);
  float inv = 1.f/(e0 + e1);
  s0[t] = e0*inv; s1[t] = e1*inv;
  __syncthreads();
  for (int st = 256; st > 0; st >>= 1){
    if (t < st){ s0[t] += s0[t+st]; s1[t] += s1[t+st]; }
    __syncthreads();
  }
  if (t == 0){ out[b*2] = s0[0]*(1.f/NPER); out[b*2+1] = s1[0]*(1.f/NPER); }
}

// ================================================================ launch
extern "C" void kernel_launch(void* const* d_in, const int* in_sizes, int n_in,
                              void* d_out, int out_size, void* d_ws, size_t ws_size,
                              hipStream_t stream)
{
  const float* nf   = (const float*)d_in[0];
  const int*   ei   = (const int*)  d_in[1];
  const int*   src  = ei;
  const int*   dst  = ei + NEDGE;
  const int*   vt   = (const int*)  d_in[2];
  const int*   ot   = (const int*)  d_in[3];
  const float* Wp   = (const float*)d_in[5];
  const float* bp   = (const float*)d_in[6];
  const float* vtab = (const float*)d_in[7];
  const float* otab = (const float*)d_in[8];
  const float* Wq   = (const float*)d_in[9];
  const float* Wk   = (const float*)d_in[10];
  const float* Wv   = (const float*)d_in[11];
  const float* bq   = (const float*)d_in[12];
  const float* bk   = (const float*)d_in[13];
  const float* bv   = (const float*)d_in[14];
  const float* Wsk  = (const float*)d_in[15];
  const float* bsk  = (const float*)d_in[16];
  const float* Wih  = (const float*)d_in[17];
  const float* Whh  = (const float*)d_in[18];
  const float* bih  = (const float*)d_in[19];
  const float* bhh  = (const float*)d_in[20];
  const float* miw  = (const float*)d_in[21];
  const float* mib  = (const float*)d_in[22];
  const float* mow  = (const float*)d_in[23];
  const float* mob  = (const float*)d_in[24];
  const float* Wc   = (const float*)d_in[25];
  const float* bc   = (const float*)d_in[26];

  char* ws = (char*)d_ws;
  float* x    = (float*)(ws);                     // 8 MB   node features x
  float* skip = (float*)(ws + ((size_t)8<<20));   // 8 MB   skip (later MHA O)
  float* agg  = (float*)(ws + ((size_t)16<<20));  // 8 MB   agg (later taint feats)
  float* Qb   = (float*)(ws + ((size_t)24<<20));  // 32 MB  Q, then V, then gi
  float* Kb   = (float*)(ws + ((size_t)56<<20));  // 32 MB  K, then MHA qkv
  float* sc   = (float*)(ws + ((size_t)88<<20));  // 2 MB   scores / ex / alpha
  float* mbuf = (float*)(ws + ((size_t)90<<20));  // 256 KB segment max
  float* den  = (float*)(ws + ((size_t)91<<20));  // 256 KB segment sum

  float* outComb = (float*)d_out;                           // [16384,128]
  float* outGE   = outComb + (size_t)NNODES*HDIM;           // [32,128]
  float* outTG   = outGE   + (size_t)NGRAPH*HDIM;           // [32,2]
  float* outTL   = outTG   + (size_t)NGRAPH*2;              // [16384,2]
  float* outUD   = outTL   + (size_t)NNODES*2;              // [16384,128]

  const dim3 blk(256);
  const dim3 gN128(HDIM/64, NNODES/128);
  const dim3 gN384(384/64,  NNODES/128);
  const dim3 gN512(512/64,  NNODES/128);

  // 1. x = nf @ Wp + bp + tiled type embeddings
  k_gemm<false><<<gN128, blk, 0, stream>>>(nf, Wp, bp, x, NNODES, HDIM, 128);
  k_add_embed<<<(NNODES*HDIM)/256, blk, 0, stream>>>(x, vt, ot, vtab, otab);

  // 2. three TransformerConv layers
  for (int l = 0; l < 3; ++l){
    const float* lWq = Wq  + (size_t)l*128*512;
    const float* lWk = Wk  + (size_t)l*128*512;
    const float* lWv = Wv  + (size_t)l*128*512;
    const float* lbq = bq  + (size_t)l*512;
    const float* lbk = bk  + (size_t)l*512;
    const float* lbv = bv  + (size_t)l*512;
    const float* lWs = Wsk + (size_t)l*128*128;
    const float* lbs = bsk + (size_t)l*128;

    k_gemm<false><<<gN512, blk, 0, stream>>>(x, lWq, lbq, Qb, NNODES, 512, 128);
    k_gemm<false><<<gN512, blk, 0, stream>>>(x, lWk, lbk, Kb, NNODES, 512, 128);
    k_fill<<<(NNODES*4)/256, blk, 0, stream>>>(mbuf, -3.4e38f, NNODES*4);
    k_fill<<<(NNODES*4)/256, blk, 0, stream>>>(den, 0.f, NNODES*4);
    k_edge_scores<<<(NEDGE*4)/256, blk, 0, stream>>>(Qb, Kb, src, dst, sc, mbuf);
    k_edge_exp   <<<(NEDGE*4)/256, blk, 0, stream>>>(dst, mbuf, sc, den);
    k_edge_alpha <<<(NEDGE*4)/256, blk, 0, stream>>>(dst, den, sc);
    k_gemm<false><<<gN512, blk, 0, stream>>>(x, lWv, lbv, Qb, NNODES, 512, 128); // V over Q
    k_fill<<<(NNODES*HDIM)/256, blk, 0, stream>>>(agg, 0.f, NNODES*HDIM);
    k_edge_agg<<<(NEDGE*HDIM)/256, blk, 0, stream>>>(src, dst, sc, Qb, agg);
    k_gemm<false><<<gN128, blk, 0, stream>>>(x, lWs, lbs, skip, NNODES, HDIM, 128);
    k_conv_finish<<<(NNODES*HDIM)/256, blk, 0, stream>>>(agg, skip, x);
  }

  // 3. GRU: gi = x @ Wih^T + bih, then sequential scan
  float* gi = Qb;
  k_gemm<true><<<gN384, blk, 0, stream>>>(x, Wih, bih, gi, NNODES, 384, 128);
  float* taint = agg;
  k_gru<<<NGRAPH, 384, 0, stream>>>(gi, Whh, bhh, taint);
  k_taint_logits<<<(NNODES*2)/256, blk, 0, stream>>>(taint, Wc, bc, outTL);

  // 4. MHA
  float* qkv = Kb;
  k_gemm<true><<<gN384, blk, 0, stream>>>(x, miw, mib, qkv, NNODES, 384, 128);
  float* Ob = skip;
  k_mha<<<dim3(NGRAPH, 4, 4), 128, 0, stream>>>(qkv, Ob);
  k_gemm<true><<<gN128, blk, 0, stream>>>(Ob, mow, mob, outUD, NNODES, HDIM, 128);

  // 5. combine + graph reductions
  k_combine<<<(NNODES*HDIM)/256, blk, 0, stream>>>(x, taint, outUD, outComb);
  k_graph_mean<<<NGRAPH, 128, 0, stream>>>(outComb, outGE);
  k_taint_graph<<<NGRAPH, 512, 0, stream>>>(outTL, outTG);

  (void)in_sizes; (void)n_in; (void)out_size; (void)ws_size;
}